// FlowProjectionModule_68891275428437
// MI455X (gfx1250) — compile-verified
//
#include <hip/hip_runtime.h>
#include <stdint.h>

// Problem dimensions (fixed by the reference).
#define Bn 16
#define Cn 2
#define Hn 720
#define Wn 1280
#define PLANE (Hn * Wn)
#define NPIX (Bn * Hn * Wn)

// Native f32 atomic add (relaxed, device scope) -> global_atomic_add_f32 on gfx1250.
__device__ __forceinline__ void atom_add_f32(float* p, float v) {
    __hip_atomic_fetch_add(p, v, __ATOMIC_RELAXED, __HIP_MEMORY_SCOPE_AGENT);
}

// ---------------------------------------------------------------------------
// Kernel 1: forward scatter. grid = (Wn/256, Hn, Bn) so (x,y,b) come straight
// from block indices (no integer div/mod chains). Valid targets add
// (-fx, -fy, 1) to the 4 corner pixels (duplicates accumulate, matching
// jnp .at[idx].add semantics).
// ---------------------------------------------------------------------------
__global__ void __launch_bounds__(256) fp_scatter(const float* __restrict__ flow,
                                                  float* __restrict__ out,
                                                  float* __restrict__ cnt) {
    const int x = blockIdx.x * 256 + threadIdx.x;
    const int y = blockIdx.y;
    const int b = blockIdx.z;

    const size_t rowoff = ((size_t)b * Hn + y) * Wn;          // within (B,H,W)
    float fx = flow[(size_t)(b * 2 + 0) * PLANE + (size_t)y * Wn + x];
    float fy = flow[(size_t)(b * 2 + 1) * PLANE + (size_t)y * Wn + x];

    float x2 = (float)x + fx;
    float y2 = (float)y + fy;
    bool valid = (x2 >= 0.0f) && (y2 >= 0.0f) &&
                 (x2 <= (float)(Wn - 1)) && (y2 <= (float)(Hn - 1));
    if (!valid) return;

    int xL = (int)floorf(x2);            // in [0, Wn-1] because valid
    int yT = (int)floorf(y2);            // in [0, Hn-1] because valid
    int xR = min(xL + 1, Wn - 1);
    int yB = min(yT + 1, Hn - 1);

    float vx = -fx;
    float vy = -fy;

    float* ox = out + (size_t)(b * 2 + 0) * PLANE;
    float* oy = out + (size_t)(b * 2 + 1) * PLANE;
    float* cb = cnt + (size_t)b * PLANE;

    int i00 = yT * Wn + xL;
    int i01 = yT * Wn + xR;
    int i10 = yB * Wn + xL;
    int i11 = yB * Wn + xR;

    atom_add_f32(&ox[i00], vx); atom_add_f32(&oy[i00], vy); atom_add_f32(&cb[i00], 1.0f);
    atom_add_f32(&ox[i01], vx); atom_add_f32(&oy[i01], vy); atom_add_f32(&cb[i01], 1.0f);
    atom_add_f32(&ox[i10], vx); atom_add_f32(&oy[i10], vy); atom_add_f32(&cb[i10], 1.0f);
    atom_add_f32(&ox[i11], vx); atom_add_f32(&oy[i11], vy); atom_add_f32(&cb[i11], 1.0f);
    (void)rowoff;
}

// ---------------------------------------------------------------------------
// Kernel 2: normalize. out = sum/cnt where cnt>0 else 0. Same 3D grid.
// ---------------------------------------------------------------------------
__global__ void __launch_bounds__(256) fp_normalize(float* __restrict__ out,
                                                    const float* __restrict__ cnt) {
    const int x = blockIdx.x * 256 + threadIdx.x;
    const int y = blockIdx.y;
    const int b = blockIdx.z;

    const size_t p = (size_t)y * Wn + x;
    float c = cnt[(size_t)b * PLANE + p];
    size_t ix = (size_t)(b * 2 + 0) * PLANE + p;
    size_t iy = (size_t)(b * 2 + 1) * PLANE + p;
    if (c > 0.0f) {
        float inv = 1.0f / c;
        out[ix] *= inv;
        out[iy] *= inv;
    } else {
        out[ix] = 0.0f;
        out[iy] = 0.0f;
    }
}

// ---------------------------------------------------------------------------
// Kernel 3: hole fill. grid = (Wn/256, Hn, Bn), block = 256 (8 waves, wave32).
// Each block stages its full cnt row (1280 floats = 5 KB) into LDS with
// CDNA5 async global->LDS copies (ASYNCcnt), then hole pixels scan LDS for
// nearest-valid left/right and L2-resident cnt columns for up/down.
// In-place is safe: writes touch only hole pixels, reads only valid pixels.
// ---------------------------------------------------------------------------
__global__ void __launch_bounds__(256) fp_fill(float* __restrict__ out,
                                               const float* __restrict__ cnt) {
    const int b = blockIdx.z;
    const int y = blockIdx.y;
    const int x = blockIdx.x * 256 + threadIdx.x;

    __shared__ float rowHas[Wn];

    const float* rowbase = cnt + ((size_t)b * Hn + y) * Wn;
    // LDS byte offset of rowHas: low 32 bits of the generic pointer are the
    // workgroup-relative LDS address (flat-LDS aperture truncation).
    unsigned lds0 = (unsigned)(uintptr_t)(void*)rowHas;

#pragma unroll
    for (int j = 0; j < Wn / 256; ++j) {
        unsigned voff = (unsigned)(threadIdx.x + j * 256) * 4u;
        unsigned ldsa = lds0 + voff;
        asm volatile("global_load_async_to_lds_b32 %0, %1, %2 offset:0"
                     :
                     : "v"(ldsa), "v"(voff), "s"(rowbase)
                     : "memory");
    }
    asm volatile("s_wait_asynccnt 0" ::: "memory");
    __syncthreads();

    if (rowHas[x] > 0.0f) return;  // not a hole

    // Nearest valid strictly left / right (LDS row scan).
    int lx = -1;
    for (int i = x - 1; i >= 0; --i) {
        if (rowHas[i] > 0.0f) { lx = i; break; }
    }
    int rx = -1;
    for (int i = x + 1; i < Wn; ++i) {
        if (rowHas[i] > 0.0f) { rx = i; break; }
    }
    // Nearest valid strictly up / down (L2-resident column scan).
    const float* colbase = cnt + (size_t)b * PLANE + x;
    int uy = -1;
    for (int i = y - 1; i >= 0; --i) {
        if (colbase[(size_t)i * Wn] > 0.0f) { uy = i; break; }
    }
    int dy = -1;
    for (int i = y + 1; i < Hn; ++i) {
        if (colbase[(size_t)i * Wn] > 0.0f) { dy = i; break; }
    }

    float num = (float)((lx >= 0) + (rx >= 0) + (uy >= 0) + (dy >= 0));
    if (num <= 0.0f) return;  // fully empty row+column: keep 0 from normalize

#pragma unroll
    for (int c = 0; c < 2; ++c) {
        const float* oc = out + (size_t)(b * 2 + c) * PLANE;
        float s = 0.0f;
        if (lx >= 0) s += oc[(size_t)y * Wn + lx];
        if (rx >= 0) s += oc[(size_t)y * Wn + rx];
        if (uy >= 0) s += oc[(size_t)uy * Wn + x];
        if (dy >= 0) s += oc[(size_t)dy * Wn + x];
        out[(size_t)(b * 2 + c) * PLANE + (size_t)y * Wn + x] = s / num;
    }
}

// ---------------------------------------------------------------------------
extern "C" void kernel_launch(void* const* d_in, const int* in_sizes, int n_in,
                              void* d_out, int out_size, void* d_ws, size_t ws_size,
                              hipStream_t stream) {
    const float* flow = (const float*)d_in[0];
    float* out = (float*)d_out;             // (B, 2, H, W) f32
    float* cnt = (float*)d_ws;              // (B, H, W) f32 scratch, 59 MB

    // Re-zero accumulators every call (graph-capture-safe stream memsets).
    hipMemsetAsync(out, 0, (size_t)Bn * Cn * Hn * Wn * sizeof(float), stream);
    hipMemsetAsync(cnt, 0, (size_t)Bn * Hn * Wn * sizeof(float), stream);

    dim3 grid(Wn / 256, Hn, Bn);            // (5, 720, 16)
    fp_scatter<<<grid, 256, 0, stream>>>(flow, out, cnt);
    fp_normalize<<<grid, 256, 0, stream>>>(out, cnt);
    fp_fill<<<grid, 256, 0, stream>>>(out, cnt);
}